// MoERouter_89996744721046
// MI455X (gfx1250) — compile-verified
//
#include <hip/hip_runtime.h>
#include <hip/hip_bf16.h>

#define HIDDEN   1024
#define NEXP     8
#define TPB      256          // threads per block (8 wave32s)
#define TOKS_PB  128          // 16 tokens per wave * 8 waves
#define WSTRIDE  1036         // LDS row stride (words) for W_gate: bank-conflict-free b64 reads
#define LSTRIDE  17           // LDS row stride (words) for logits tile

typedef __attribute__((ext_vector_type(2))) float v2f;
typedef __attribute__((ext_vector_type(8))) float v8f;

__global__ void moe_router_init(float* ws) {
    if (threadIdx.x < 2 * NEXP) ws[threadIdx.x] = 0.0f;
}

__global__ __launch_bounds__(TPB) void moe_router_main(
    const float* __restrict__ H,      // [T, 1024]
    const float* __restrict__ Wg,     // [8, 1024]
    float* __restrict__ out_w,        // [T, 2]
    float* __restrict__ out_sel,      // [T, 2] (indices as float)
    float* __restrict__ ws)           // [0..7]=prob sums, [8..15]=sel counts
{
    __shared__ float ldsW[9 * WSTRIDE];            // 8 expert rows + 1 zero row
    __shared__ float ldsL[TOKS_PB * LSTRIDE];      // logits tile
    __shared__ float sPsum[NEXP];
    __shared__ float sCnt[NEXP];

    const int tid  = threadIdx.x;
    const int lane = tid & 31;
    const int wave = tid >> 5;

    // ---- Phase A: stage W_gate into LDS (coalesced b128), zero pad row ----
    for (int g = tid * 4; g < NEXP * HIDDEN; g += TPB * 4) {
        const int e  = g >> 10;
        const int kk = g & (HIDDEN - 1);
        const float4 v = *reinterpret_cast<const float4*>(Wg + g);
        *reinterpret_cast<float4*>(&ldsW[e * WSTRIDE + kk]) = v;
    }
    if (tid * 4 < HIDDEN) {
        float4 z; z.x = 0.f; z.y = 0.f; z.z = 0.f; z.w = 0.f;
        *reinterpret_cast<float4*>(&ldsW[8 * WSTRIDE + tid * 4]) = z;
    }
    if (tid < NEXP) { sPsum[tid] = 0.0f; sCnt[tid] = 0.0f; }
    __syncthreads();

    // ---- Phase B: gate GEMM via V_WMMA_F32_16X16X4_F32 ----
    // A (16x4 f32): lanes 0-15 hold M=lane, K={0,1}; lanes 16-31 hold M=lane-16, K={2,3}
    // B (4x16 f32): lanes 0-15 hold N=lane, K={0,1}; lanes 16-31 hold N=lane-16, K={2,3}
    const int m    = lane & 15;
    const int hi   = lane >> 4;
    const int koff = hi * 2;
    const int n    = lane & 15;
    const int nw   = (n < NEXP) ? n : NEXP;        // padding lanes read the zero row

    const int tok0 = blockIdx.x * TOKS_PB + wave * 16;
    const float2* hp = reinterpret_cast<const float2*>(
        H + (size_t)(tok0 + m) * HIDDEN + koff);
    const float2* bp = reinterpret_cast<const float2*>(
        &ldsW[nw * WSTRIDE + koff]);

    v8f c = {};
    #pragma unroll 8
    for (int k = 0; k < HIDDEN; k += 4) {
        const float2 av = hp[k >> 1];              // global_load_b64
        const float2 bv = bp[k >> 1];              // ds_load_b64
        v2f a; a.x = av.x; a.y = av.y;
        v2f b; b.x = bv.x; b.y = bv.y;
        c = __builtin_amdgcn_wmma_f32_16x16x4_f32(
                /*neg_a=*/false, a, /*neg_b=*/false, b,
                /*c_mod=*/(short)0, c, /*reuse_a=*/false, /*reuse_b=*/false);
    }

    // Scatter C tile: VGPR r holds (row r, lanes 0-15) and (row r+8, lanes 16-31)
    if (n < NEXP) {
        #pragma unroll
        for (int r = 0; r < 8; ++r) {
            const int mt = wave * 16 + r + 8 * hi;
            ldsL[mt * LSTRIDE + n] = c[r];
        }
    }
    __syncthreads();

    // ---- Phase C: softmax over 8 experts, top-2, outputs, aux accumulation ----
    if (tid < TOKS_PB) {
        float p[NEXP];
        float mx = -1e30f;
        #pragma unroll
        for (int e = 0; e < NEXP; ++e) {
            p[e] = ldsL[tid * LSTRIDE + e];
            mx = fmaxf(mx, p[e]);
        }
        float s = 0.0f;
        #pragma unroll
        for (int e = 0; e < NEXP; ++e) { p[e] = expf(p[e] - mx); s += p[e]; }
        const float inv = 1.0f / s;
        #pragma unroll
        for (int e = 0; e < NEXP; ++e) {
            p[e] *= inv;
            atomicAdd(&sPsum[e], p[e]);
        }

        // top-2, lowest index wins ties (strict > keeps first occurrence)
        int i1 = 0; float p1 = p[0];
        #pragma unroll
        for (int e = 1; e < NEXP; ++e) if (p[e] > p1) { p1 = p[e]; i1 = e; }
        int i2 = (i1 == 0) ? 1 : 0; float p2 = p[i2];
        #pragma unroll
        for (int e = 0; e < NEXP; ++e)
            if (e != i1 && p[e] > p2) { p2 = p[e]; i2 = e; }

        const float rs = 1.0f / (p1 + p2);
        const int gt = blockIdx.x * TOKS_PB + tid;
        out_w[2 * gt + 0]   = p1 * rs;
        out_w[2 * gt + 1]   = p2 * rs;
        out_sel[2 * gt + 0] = (float)i1;
        out_sel[2 * gt + 1] = (float)i2;

        atomicAdd(&sCnt[i1], 1.0f);
        atomicAdd(&sCnt[i2], 1.0f);
    }
    __syncthreads();

    if (tid < NEXP) {
        atomicAdd(&ws[tid],        sPsum[tid]);
        atomicAdd(&ws[NEXP + tid], sCnt[tid]);
    }
}

__global__ void moe_router_finalize(const float* __restrict__ ws,
                                    float* __restrict__ aux, float invT) {
    if (threadIdx.x == 0) {
        float s = 0.0f;
        #pragma unroll
        for (int e = 0; e < NEXP; ++e) s += ws[e] * ws[NEXP + e];
        aux[0] = (float)NEXP * s * invT * invT;
    }
}

extern "C" void kernel_launch(void* const* d_in, const int* in_sizes, int n_in,
                              void* d_out, int out_size, void* d_ws, size_t ws_size,
                              hipStream_t stream) {
    const float* H  = (const float*)d_in[0];   // [4,8192,1024] fp32
    const float* Wg = (const float*)d_in[1];   // [8,1024] fp32
    float* out = (float*)d_out;
    float* ws  = (float*)d_ws;

    const int nTok = in_sizes[0] / HIDDEN;     // 32768
    float* out_w   = out;                      // [T,2]
    float* out_sel = out + 2 * nTok;           // [T,2]
    float* out_aux = out + 4 * nTok;           // [1]

    moe_router_init<<<1, 32, 0, stream>>>(ws);
    moe_router_main<<<nTok / TOKS_PB, TPB, 0, stream>>>(H, Wg, out_w, out_sel, ws);
    moe_router_finalize<<<1, 32, 0, stream>>>(ws, out_aux, 1.0f / (float)nTok);
}